// UVNetGraphEncoder_86517821216288
// MI455X (gfx1250) — compile-verified
//
#include <hip/hip_runtime.h>

#define NN 60000
#define NE 300000

typedef float v2f __attribute__((ext_vector_type(2)));
typedef float v8f __attribute__((ext_vector_type(8)));

__device__ __forceinline__ v8f wmma_f32(v2f a, v2f b, v8f c) {
  // D = A(16x4,f32) * B(4x16,f32) + C(16x16,f32)
  return __builtin_amdgcn_wmma_f32_16x16x4_f32(false, a, false, b, (short)0, c, false, false);
}

// LDS byte offset of a generic pointer to __shared__ (LDS aperture: addr[31:0])
__device__ __forceinline__ unsigned lds_off(const void* p) {
  return (unsigned)(size_t)p;
}

// async 16B global->LDS copy, uniform 64-bit base in SGPRs + per-lane byte offset
__device__ __forceinline__ void async_cp16_s(void* lds, const void* gbase, int byte_off) {
  asm volatile("global_load_async_to_lds_b128 %0, %1, %2"
               :: "v"(lds_off(lds)), "v"(byte_off), "s"(gbase) : "memory");
}
// async 16B global->LDS copy, per-lane 64-bit global address
__device__ __forceinline__ void async_cp16_v(void* lds, const void* gaddr) {
  asm volatile("global_load_async_to_lds_b128 %0, %1, off"
               :: "v"(lds_off(lds)), "v"(gaddr) : "memory");
}
__device__ __forceinline__ void wait_async0() {
  asm volatile("s_wait_asynccnt 0x0" ::: "memory");
}

// order-preserving float<->uint encoding so atomicMax(uint) == float max; 0 is identity
__device__ __forceinline__ unsigned enc_f(float x) {
  unsigned u = __float_as_uint(x);
  return (u & 0x80000000u) ? ~u : (u | 0x80000000u);
}
__device__ __forceinline__ float dec_f(unsigned u) {
  return (u & 0x80000000u) ? __uint_as_float(u & 0x7FFFFFFFu) : __uint_as_float(~u);
}

__device__ __forceinline__ float wave_sum(float v) {
#pragma unroll
  for (int o = 16; o > 0; o >>= 1) v += __shfl_xor(v, o, 32);
  return v;
}

// ---------------------------------------------------------------------------
// Y[rows x 64] = X[rows x K] @ W[K x 64] (+ bias).  K in {16,32,64}, K%4==0.
// One 16-row tile per wave, W staged in LDS via async-to-LDS DMA, f32 WMMA.
// ---------------------------------------------------------------------------
__global__ void gemm_wmma64(const float* __restrict__ X, const float* __restrict__ W,
                            const float* __restrict__ bias, float* __restrict__ Y,
                            int rows, int K) {
  __shared__ float lW[64 * 64];
  __shared__ float lB[64];
  const int F = 64;
  int tid = threadIdx.x;
  for (int i = tid * 4; i < K * F; i += blockDim.x * 4)
    async_cp16_s(&lW[i], W, i * 4);
  if (tid < F) lB[tid] = bias ? bias[tid] : 0.f;
  wait_async0();
  __syncthreads();

  int wave = tid >> 5, lane = tid & 31;
  int row0 = (blockIdx.x * (blockDim.x >> 5) + wave) * 16;
  if (row0 >= rows) return;
  int r = lane & 15, hi = lane >> 4;
  const float* xr = X + (size_t)(row0 + r) * K;

  v8f acc[4] = {};
  for (int s = 0; s < (K >> 2); ++s) {
    int k0 = 4 * s + hi * 2;
    v2f a;
    a.x = xr[k0];
    a.y = xr[k0 + 1];
#pragma unroll
    for (int t = 0; t < 4; ++t) {
      v2f b;
      b.x = lW[k0 * F + t * 16 + r];
      b.y = lW[(k0 + 1) * F + t * 16 + r];
      acc[t] = wmma_f32(a, b, acc[t]);
    }
  }
#pragma unroll
  for (int t = 0; t < 4; ++t) {
    int col = t * 16 + r;
    float bv = lB[col];
#pragma unroll
    for (int m = 0; m < 8; ++m) {
      int rr = m + hi * 8;
      Y[(size_t)(row0 + rr) * F + col] = acc[t][m] + bv;
    }
  }
}

// ---------------------------------------------------------------------------
// EGAT edge combine: f_out = leaky(hi[src] + hj[dst] + f@Wfij(+bias, folded))
// plus attention logit + running segment-max (encoded atomicMax). Wave/edge.
// ---------------------------------------------------------------------------
__global__ void egat_combine(const float* __restrict__ hi_, const float* __restrict__ hj_,
                             const float* __restrict__ fin, const float* __restrict__ attn,
                             const int* __restrict__ src, const int* __restrict__ dst,
                             float* __restrict__ fout, float* __restrict__ logits,
                             unsigned* __restrict__ emax, int E) {
  int wave = threadIdx.x >> 5, lane = threadIdx.x & 31;
  int e = blockIdx.x * (blockDim.x >> 5) + wave;
  if (e >= E) return;
  int s = src[e], d = dst[e];
  size_t eb = (size_t)e * 64;
  float v0 = hi_[(size_t)s * 64 + lane] + hj_[(size_t)d * 64 + lane] + fin[eb + lane];
  float v1 = hi_[(size_t)s * 64 + 32 + lane] + hj_[(size_t)d * 64 + 32 + lane] + fin[eb + 32 + lane];
  v0 = v0 >= 0.f ? v0 : 0.01f * v0;
  v1 = v1 >= 0.f ? v1 : 0.01f * v1;
  if (fout) { fout[eb + lane] = v0; fout[eb + 32 + lane] = v1; }
  float p = v0 * attn[lane] + v1 * attn[32 + lane];
  p = wave_sum(p);
  if (lane == 0) {
    logits[e] = p;
    atomicMax(&emax[d], enc_f(p));
  }
}

__global__ void edge_exp(const float* __restrict__ logits, const unsigned* __restrict__ emax,
                         const int* __restrict__ dst, float* __restrict__ eexp,
                         float* __restrict__ denom, int E) {
  int i = blockIdx.x * blockDim.x + threadIdx.x;
  if (i >= E) return;
  int d = dst[i];
  float e = __expf(logits[i] - dec_f(emax[d]));
  eexp[i] = e;
  atomicAdd(&denom[d], e);
}

__global__ void egat_aggregate(const float* __restrict__ hn, const int* __restrict__ src,
                               const int* __restrict__ dst, const float* __restrict__ eexp,
                               const float* __restrict__ denom, float* __restrict__ hout, int E) {
  long long idx = (long long)blockIdx.x * blockDim.x + threadIdx.x;
  if (idx >= (long long)E * 64) return;
  int e = (int)(idx >> 6), ch = (int)(idx & 63);
  int d = dst[e];
  float a = eexp[e] / denom[d];
  atomicAdd(&hout[(size_t)d * 64 + ch], hn[(size_t)src[e] * 64 + ch] * a);
}

__global__ void count_edges(const int* __restrict__ dst, float* __restrict__ cnt, int E) {
  int i = blockIdx.x * blockDim.x + threadIdx.x;
  if (i < E) atomicAdd(&cnt[dst[i]], 1.f);
}

// ---------------------------------------------------------------------------
// NNConv: m[e,o] = sum_{k,i} ef[e,k]*x[src[e],i]*nn_w[k, i*32+o]  (+ x@nn_b)
// = GEMM with virtual A  v[e, k*32+i] = ef[e,k]*x_src[e,i]  (K=512) against
// nn_w reinterpreted as 512x32 (the reshape is the identity:
// (k*32+i)*32+o == k*1024+i*32+o), held in LDS via async-to-LDS DMA.
// 16-edge tile per wave; scatter-add to dst.
// ---------------------------------------------------------------------------
#define NN_WAVES 4
__global__ void nnconv_wmma(const float* __restrict__ face, const float* __restrict__ ef,
                            const float* __restrict__ nnw, const float* __restrict__ nnb,
                            const int* __restrict__ src, const int* __restrict__ dst,
                            float* __restrict__ sums, int E) {
  extern __shared__ float sm[];
  float* lW  = sm;                           // 512*32   nn_w (flat copy)
  float* lBb = lW + 512 * 32;                // 32*32    nn_b as matrix
  float* sx  = lBb + 1024;                   // NN_WAVES*16*32  gathered x_src
  float* sef = sx + NN_WAVES * 512;          // NN_WAVES*16*16  edge feats
  int*   sdst = (int*)(sef + NN_WAVES * 256);// NN_WAVES*16

  int tid = threadIdx.x;
  for (int i = tid * 4; i < 512 * 32; i += blockDim.x * 4)
    async_cp16_s(&lW[i], nnw, i * 4);
  for (int i = tid * 4; i < 1024; i += blockDim.x * 4)
    async_cp16_s(&lBb[i], nnb, i * 4);

  int wave = tid >> 5, lane = tid & 31;
  int tile = blockIdx.x * NN_WAVES + wave;
  int e0 = tile * 16;
  bool valid = e0 < E;
  int wx = wave * 512, wo = wave * 256;
  if (valid) {
    // 16 rows of 32 floats (128B each): 4x 16B per lane, row-contiguous
    for (int t = lane * 4; t < 16 * 32; t += 32 * 4) {
      int el = t >> 5, ch = t & 31;
      async_cp16_v(&sx[wx + t], face + (size_t)src[e0 + el] * 32 + ch);
    }
    // 16 rows of 16 floats (64B each): 2x 16B per lane
    for (int t = lane * 4; t < 16 * 16; t += 32 * 4) {
      int el = t >> 4, ch = t & 15;
      async_cp16_v(&sef[wo + t], ef + (size_t)(e0 + el) * 16 + ch);
    }
    if (lane < 16) sdst[wave * 16 + lane] = dst[e0 + lane];
  }
  wait_async0();
  __syncthreads();
  if (!valid) return;

  int r = lane & 15, hi = lane >> 4;
  v8f acc0 = {}, acc1 = {};
  for (int s = 0; s < 128; ++s) {            // K = 512
    int k0 = 4 * s + hi * 2, k1 = k0 + 1;
    v2f a;
    a.x = sef[wo + r * 16 + (k0 >> 5)] * sx[wx + r * 32 + (k0 & 31)];
    a.y = sef[wo + r * 16 + (k1 >> 5)] * sx[wx + r * 32 + (k1 & 31)];
    v2f b0, b1;
    b0.x = lW[k0 * 32 + r];      b0.y = lW[k1 * 32 + r];
    b1.x = lW[k0 * 32 + 16 + r]; b1.y = lW[k1 * 32 + 16 + r];
    acc0 = wmma_f32(a, b0, acc0);
    acc1 = wmma_f32(a, b1, acc1);
  }
  for (int s = 0; s < 8; ++s) {              // + x_src @ reshape(nn_b,32,32)
    int k0 = 4 * s + hi * 2, k1 = k0 + 1;
    v2f a;
    a.x = sx[wx + r * 32 + k0];
    a.y = sx[wx + r * 32 + k1];
    v2f b0, b1;
    b0.x = lBb[k0 * 32 + r];      b0.y = lBb[k1 * 32 + r];
    b1.x = lBb[k0 * 32 + 16 + r]; b1.y = lBb[k1 * 32 + 16 + r];
    acc0 = wmma_f32(a, b0, acc0);
    acc1 = wmma_f32(a, b1, acc1);
  }
#pragma unroll
  for (int m = 0; m < 8; ++m) {
    int rr = m + hi * 8;
    int d = sdst[wave * 16 + rr];
    atomicAdd(&sums[(size_t)d * 32 + r], acc0[m]);
    atomicAdd(&sums[(size_t)d * 32 + 16 + r], acc1[m]);
  }
}

// ---------------------------------------------------------------------------
// Assemble node_features = [Gf | Ef | 0], gate logit + global max. Wave/node.
// ---------------------------------------------------------------------------
__global__ void assemble_gate(const float* __restrict__ h2, const float* __restrict__ sums,
                              const float* __restrict__ cnt, const float* __restrict__ nn_bias,
                              const float* __restrict__ gate_w, const float* __restrict__ gate_b,
                              float* __restrict__ out, float* __restrict__ glog,
                              unsigned* __restrict__ gmax, int n) {
  int wave = threadIdx.x >> 5, lane = threadIdx.x & 31;
  int node = blockIdx.x * (blockDim.x >> 5) + wave;
  if (node >= n) return;
  float c = fmaxf(cnt[node], 1.0f);
  float p = 0.f;
#pragma unroll
  for (int q = 0; q < 4; ++q) {
    int ch = q * 32 + lane;
    float v;
    if (ch < 64)      v = h2[(size_t)node * 64 + ch];
    else if (ch < 96) v = sums[(size_t)node * 32 + (ch - 64)] / c + nn_bias[ch - 64];
    else              v = 0.f;
    out[(size_t)node * 128 + ch] = v;
    p += v * gate_w[ch];
  }
  p = wave_sum(p);
  if (lane == 0) {
    p += gate_b[0];
    glog[node] = p;
    atomicMax(gmax, enc_f(p));
  }
}

__global__ void gate_exp(const float* __restrict__ glog, const unsigned* __restrict__ gmax,
                         float* __restrict__ ge, float* __restrict__ Z, int n) {
  int i = blockIdx.x * blockDim.x + threadIdx.x;
  if (i >= n) return;
  float e = __expf(glog[i] - dec_f(gmax[0]));
  ge[i] = e;
  atomicAdd(Z, e);
}

#define POOL_NODES 256
__global__ void pool_sum(const float* __restrict__ feat, const float* __restrict__ ge,
                         const float* __restrict__ Z, float* __restrict__ g, int n) {
  int ch = threadIdx.x; // 128 threads = channels
  int start = blockIdx.x * POOL_NODES;
  int end = min(start + POOL_NODES, n);
  float acc = 0.f;
  for (int node = start; node < end; ++node)
    acc += ge[node] * feat[(size_t)node * 128 + ch];
  atomicAdd(&g[ch], acc / Z[0]);
}

// ---------------------------------------------------------------------------
extern "C" void kernel_launch(void* const* d_in, const int* in_sizes, int n_in,
                              void* d_out, int out_size, void* d_ws, size_t ws_size,
                              hipStream_t stream) {
  const float* face    = (const float*)d_in[0];
  const float* edgef   = (const float*)d_in[1];
  const float* fp_w    = (const float*)d_in[2];
  const float* fp_b    = (const float*)d_in[3];
  const float* ep_w    = (const float*)d_in[4];
  const float* ep_b    = (const float*)d_in[5];
  const float* nn_w    = (const float*)d_in[6];
  const float* nn_b    = (const float*)d_in[7];
  const float* nn_bias = (const float*)d_in[8];
  const float* gate_w  = (const float*)d_in[9];
  const float* gate_b  = (const float*)d_in[10];
  const int*   src     = (const int*)d_in[11];
  const int*   dst     = (const int*)d_in[12];
  const float* e_ni[2]   = {(const float*)d_in[13], (const float*)d_in[19]};
  const float* e_nj[2]   = {(const float*)d_in[14], (const float*)d_in[20]};
  const float* e_fij[2]  = {(const float*)d_in[15], (const float*)d_in[21]};
  const float* e_node[2] = {(const float*)d_in[16], (const float*)d_in[22]};
  const float* e_attn[2] = {(const float*)d_in[17], (const float*)d_in[23]};
  const float* e_bias[2] = {(const float*)d_in[18], (const float*)d_in[24]};

  const size_t N = NN, E = NE;
  float* out = (float*)d_out;

  // workspace layout
  float* p = (float*)d_ws;
  float* h0     = p; p += N * 64;
  float* bufA   = p; p += E * 64;   // fij out / f1 (in place)
  float* bufB   = p; p += E * 64;   // f0 / fij2 out
  float* t_hi   = p; p += N * 64;
  float* t_hj   = p; p += N * 64;
  float* t_hn   = p; p += N * 64;
  float* h1     = p; p += N * 64;   // h1,h2 contiguous (one memset)
  float* h2     = p; p += N * 64;
  float* logits = p; p += E;
  float* eexp   = p; p += E;
  unsigned* emax = (unsigned*)p; p += N;  // emax,denom contiguous
  float* denom  = p; p += N;
  float* sums   = p; p += N * 32;   // sums,cnt contiguous
  float* cnt    = p; p += N;
  float* glog   = p; p += N;
  float* ge     = p; p += N;
  unsigned* gmax = (unsigned*)p; p += 1;  // gmax,Z contiguous
  float* Zp     = p; p += 1;

  // zero accumulators (0 is identity for encoded max too)
  hipMemsetAsync(h1, 0, N * 64 * 2 * sizeof(float), stream);
  hipMemsetAsync(sums, 0, N * 33 * sizeof(float), stream);
  hipMemsetAsync(gmax, 0, 2 * sizeof(float), stream);
  hipMemsetAsync(out + N * 128, 0, 128 * sizeof(float), stream);

  const int GB = 256;                         // gemm block: 8 waves = 128 rows
  auto gemm = [&](const float* X, const float* W, const float* b, float* Y, int rows, int K) {
    gemm_wmma64<<<(rows + 127) / 128, GB, 0, stream>>>(X, W, b, Y, rows, K);
  };

  // projections
  gemm(face, fp_w, fp_b, h0, (int)N, 32);
  gemm(edgef, ep_w, ep_b, bufB, (int)E, 16);

  float* hcur = h0;
  float* fin  = bufB;
  float* ftmp = bufA;
  for (int L = 0; L < 2; ++L) {
    gemm(hcur, e_ni[L],   nullptr,   t_hi, (int)N, 64);
    gemm(hcur, e_nj[L],   nullptr,   t_hj, (int)N, 64);
    gemm(hcur, e_node[L], nullptr,   t_hn, (int)N, 64);
    gemm(fin,  e_fij[L],  e_bias[L], ftmp, (int)E, 64);   // bias folded here

    hipMemsetAsync(emax, 0, 2 * N * sizeof(float), stream);
    egat_combine<<<(NE + 7) / 8, 256, 0, stream>>>(
        t_hi, t_hj, ftmp, e_attn[L], src, dst,
        (L == 0 ? ftmp : nullptr), logits, emax, (int)E);
    edge_exp<<<(NE + 255) / 256, 256, 0, stream>>>(logits, emax, dst, eexp, denom, (int)E);
    float* hout = (L == 0) ? h1 : h2;
    egat_aggregate<<<(int)((E * 64 + 255) / 256), 256, 0, stream>>>(
        t_hn, src, dst, eexp, denom, hout, (int)E);

    hcur = hout;
    float* t = fin; fin = ftmp; ftmp = t;
  }

  // NNConv (mean aggregation)
  count_edges<<<(NE + 255) / 256, 256, 0, stream>>>(dst, cnt, (int)E);
  {
    int tiles = (int)(E / 16);
    int grid = (tiles + NN_WAVES - 1) / NN_WAVES;
    size_t smem = (512 * 32 + 1024 + NN_WAVES * (512 + 256 + 16)) * sizeof(float);
    nnconv_wmma<<<grid, NN_WAVES * 32, smem, stream>>>(face, edgef, nn_w, nn_b, src, dst,
                                                       sums, (int)E);
  }

  // concat + global attention pooling
  assemble_gate<<<(NN + 7) / 8, 256, 0, stream>>>(h2, sums, cnt, nn_bias, gate_w, gate_b,
                                                  out, glog, gmax, (int)N);
  gate_exp<<<(NN + 255) / 256, 256, 0, stream>>>(glog, gmax, ge, Zp, (int)N);
  pool_sum<<<(NN + POOL_NODES - 1) / POOL_NODES, 128, 0, stream>>>(out, ge, Zp,
                                                                   out + N * 128, (int)N);
}